// MPLayer_50328426774758
// MI455X (gfx1250) — compile-verified
//
#include <hip/hip_runtime.h>

typedef __attribute__((ext_vector_type(2))) float v2f;
typedef __attribute__((ext_vector_type(8))) float v8f;

// ---------------------------------------------------------------------------
// GEMM: Y[M x 256] = act(X[M x 256] @ W^T + bias),  W is [256 x 256] row-major
// (out[m][n] = sum_k X[m][k] * W[n][k] + bias[n]).
// One wave32 computes a 16x64 output strip with V_WMMA_F32_16X16X4_F32:
//   A tile 16x4 f32 : lane l -> row (l&15), holds K = kb, kb+1, kb=(l>>4)*2
//   B tile 4x16 f32 : lane l -> col (l&15), same K parity layout (symmetric)
//   C/D 16x16 f32   : VGPR r -> M=r (lanes 0-15) / M=8+r (lanes 16-31), N=lane&15
// ---------------------------------------------------------------------------
template <bool RELU>
__global__ void mp_gemm_wmma_f32(const float* __restrict__ X,
                                 const float* __restrict__ W,
                                 const float* __restrict__ bias,
                                 float* __restrict__ Y,
                                 int Mrows) {
  constexpr int K = 256;
  const int lane   = threadIdx.x & 31;
  const int wvInBk = threadIdx.x >> 5;
  const int gwave  = blockIdx.x * (blockDim.x >> 5) + wvInBk;

  const int mTiles = (Mrows + 15) >> 4;
  const int nQuads = 4;                      // 256 cols / 64 per wave
  if (gwave >= mTiles * nQuads) return;      // wave-uniform: EXEC stays all-1s

  const int mt = gwave / nQuads;
  const int nq = gwave - mt * nQuads;
  const int m0 = mt << 4;
  const int n0 = nq << 6;

  const int mr = m0 + (lane & 15);
  const int am = (mr < Mrows) ? mr : (Mrows - 1);   // clamp A loads, store guarded
  const int kb = (lane >> 4) << 1;                  // 0 or 2

  const float* xrow  = X + (size_t)am * K;
  const float* wrow0 = W + (size_t)(n0 +  0 + (lane & 15)) * K;
  const float* wrow1 = W + (size_t)(n0 + 16 + (lane & 15)) * K;
  const float* wrow2 = W + (size_t)(n0 + 32 + (lane & 15)) * K;
  const float* wrow3 = W + (size_t)(n0 + 48 + (lane & 15)) * K;

  v8f c0 = {}, c1 = {}, c2 = {}, c3 = {};

#pragma unroll 4
  for (int k0 = 0; k0 < K; k0 += 4) {
    const int idx = k0 + kb;
    v2f a  = *(const v2f*)(xrow  + idx);
    v2f b0 = *(const v2f*)(wrow0 + idx);
    v2f b1 = *(const v2f*)(wrow1 + idx);
    v2f b2 = *(const v2f*)(wrow2 + idx);
    v2f b3 = *(const v2f*)(wrow3 + idx);
    c0 = __builtin_amdgcn_wmma_f32_16x16x4_f32(false, a, false, b0, (short)0, c0, false, false);
    c1 = __builtin_amdgcn_wmma_f32_16x16x4_f32(false, a, false, b1, (short)0, c1, false, false);
    c2 = __builtin_amdgcn_wmma_f32_16x16x4_f32(false, a, false, b2, (short)0, c2, false, false);
    c3 = __builtin_amdgcn_wmma_f32_16x16x4_f32(false, a, false, b3, (short)0, c3, false, false);
  }

  const int cb  = lane & 15;
  const float b0v = bias[n0 +  0 + cb];
  const float b1v = bias[n0 + 16 + cb];
  const float b2v = bias[n0 + 32 + cb];
  const float b3v = bias[n0 + 48 + cb];
  const int rOff = (lane < 16) ? 0 : 8;

#pragma unroll
  for (int r = 0; r < 8; ++r) {
    const int row = m0 + rOff + r;
    if (row < Mrows) {
      float* yr = Y + (size_t)row * 256 + n0 + cb;
      float v0 = c0[r] + b0v;
      float v1 = c1[r] + b1v;
      float v2 = c2[r] + b2v;
      float v3 = c3[r] + b3v;
      if (RELU) {
        v0 = fmaxf(v0, 0.f); v1 = fmaxf(v1, 0.f);
        v2 = fmaxf(v2, 0.f); v3 = fmaxf(v3, 0.f);
      }
      yr[0] = v0; yr[16] = v1; yr[32] = v2; yr[48] = v3;
    }
  }
}

// ---------------------------------------------------------------------------
// Zero the aggregation buffer (can't hipMemset inside graph capture safely).
// ---------------------------------------------------------------------------
__global__ void mp_zero_f4(float4* __restrict__ p, int n4) {
  int i = blockIdx.x * blockDim.x + threadIdx.x;
  if (i < n4) p[i] = make_float4(0.f, 0.f, 0.f, 0.f);
}

// ---------------------------------------------------------------------------
// Scatter: agg[dst[e]] += M[src[e]]  (256 f32 channels per edge).
// One wave per edge: 2 coalesced float4 loads/lane, 8 GLOBAL_ATOMIC_ADD_F32.
// Whole M/agg working set is L2-resident (20 MB << 192 MB L2).
// ---------------------------------------------------------------------------
__global__ void mp_scatter_add(const float* __restrict__ Msg,
                               const int* __restrict__ src,
                               const int* __restrict__ dst,
                               float* __restrict__ agg,
                               int nEdges) {
  const int wave = (blockIdx.x * blockDim.x + threadIdx.x) >> 5;
  const int lane = threadIdx.x & 31;
  if (wave >= nEdges) return;

  const int s = src[wave];
  const int d = dst[wave];
  const float* mrow = Msg + (size_t)s * 256;
  float*       arow = agg + (size_t)d * 256;

  const int c0 = lane * 4;        // channels [0,128) in float4s
  const int c1 = c0 + 128;        // channels [128,256)
  float4 v0 = *(const float4*)(mrow + c0);
  float4 v1 = *(const float4*)(mrow + c1);

  atomicAdd(arow + c0 + 0, v0.x);
  atomicAdd(arow + c0 + 1, v0.y);
  atomicAdd(arow + c0 + 2, v0.z);
  atomicAdd(arow + c0 + 3, v0.w);
  atomicAdd(arow + c1 + 0, v1.x);
  atomicAdd(arow + c1 + 1, v1.y);
  atomicAdd(arow + c1 + 2, v1.z);
  atomicAdd(arow + c1 + 3, v1.w);
}

extern "C" void kernel_launch(void* const* d_in, const int* in_sizes, int n_in,
                              void* d_out, int out_size, void* d_ws, size_t ws_size,
                              hipStream_t stream) {
  const float* X   = (const float*)d_in[0];  // node_feats [N, 256]
  const int*   src = (const int*)  d_in[1];  // [E]
  const int*   dst = (const int*)  d_in[2];  // [E]
  const float* Wm  = (const float*)d_in[3];  // [256, 256]
  const float* bm  = (const float*)d_in[4];  // [256]
  const float* Wf  = (const float*)d_in[5];  // [256, 256]
  const float* bf  = (const float*)d_in[6];  // [256]
  float*       out = (float*)d_out;          // [N, 256]

  const int nNodes = in_sizes[0] / 256;      // 10000
  const int nEdges = in_sizes[1];            // 320000

  float* Mbuf = (float*)d_ws;                         // [N, 256]
  float* Agg  = Mbuf + (size_t)nNodes * 256;          // [N, 256]

  // GEMM launch geometry: (mTiles * 4) waves, 8 waves (256 threads) per block.
  const int mTiles    = (nNodes + 15) / 16;
  const int gemmWaves = mTiles * 4;
  const int gemmBlks  = (gemmWaves + 7) / 8;

  // 1) M = relu(X @ Wm^T + bm)
  mp_gemm_wmma_f32<true><<<gemmBlks, 256, 0, stream>>>(X, Wm, bm, Mbuf, nNodes);

  // 2) Agg = 0
  const int n4 = nNodes * 256 / 4;
  mp_zero_f4<<<(n4 + 255) / 256, 256, 0, stream>>>((float4*)Agg, n4);

  // 3) Agg[dst] += M[src]  (atomic f32 adds at L2)
  const long long scatterThreads = (long long)nEdges * 32;
  mp_scatter_add<<<(int)((scatterThreads + 255) / 256), 256, 0, stream>>>(
      Mbuf, src, dst, Agg, nEdges);

  // 4) out = Agg @ Wf^T + bf
  mp_gemm_wmma_f32<false><<<gemmBlks, 256, 0, stream>>>(Agg, Wf, bf, out, nNodes);
}